// AvULoss_15573551415398
// MI455X (gfx1250) — compile-verified
//
#include <hip/hip_runtime.h>
#include <math.h>

typedef float f32x4 __attribute__((ext_vector_type(4)));
typedef int   i32x4 __attribute__((ext_vector_type(4)));
typedef float v2f   __attribute__((ext_vector_type(2)));
typedef float v8f   __attribute__((ext_vector_type(8)));

#define AVU_EPS 1e-10f

#if __has_builtin(__builtin_amdgcn_tanhf)
#define FAST_TANH(x) __builtin_amdgcn_tanhf(x)
#else
#define FAST_TANH(x) tanhf(x)
#endif

// Wave32 sum reduction via V_WMMA_F32_16X16X4_F32.
// A (16x4) layout: VGPR0 lanes0-15 -> K=0 (M=lane), lanes16-31 -> K=2 (M=lane-16);
//                  VGPR1 lanes0-15 -> K=1,         lanes16-31 -> K=3.
// With A = {v, 0} per lane and B = all-ones (layout independent), every
// D[m][n] = v(m) + v(m+16). D VGPR r holds M=r (lanes 0-15) / M=8+r (lanes 16-31),
// so summing the 8 D VGPRs gives sum over half the lanes; one xor-16 shuffle
// completes the 32-lane sum (replicated to all lanes).
__device__ __forceinline__ float wave_reduce_sum_wmma(float v) {
    v2f a; a.x = v;    a.y = 0.0f;
    v2f b; b.x = 1.0f; b.y = 1.0f;
    v8f c = {};
    v8f d = __builtin_amdgcn_wmma_f32_16x16x4_f32(
        /*neg_a=*/false, a, /*neg_b=*/false, b,
        /*c_mod=*/(short)0, c, /*reuse_a=*/false, /*reuse_b=*/false);
    float s = ((d[0] + d[1]) + (d[2] + d[3])) + ((d[4] + d[5]) + (d[6] + d[7]));
    s += __shfl_xor(s, 16, 32);
    return s;
}

__global__ void avu_ws_init_kernel(float* __restrict__ ws) {
    ws[0] = 0.0f;
    ws[1] = 0.0f;
}

__global__ void __launch_bounds__(256)
avu_partial_kernel(const float* __restrict__ probs,
                   const int*   __restrict__ labels,
                   const float* __restrict__ unc,
                   const float* __restrict__ unc_th,
                   float* __restrict__ ws,
                   int n) {
    const float th = unc_th[0];

    float num = 0.0f;  // n_ac + n_iu
    float den = 0.0f;  // n_ac + n_au + n_ic + n_iu

    const int tid      = blockIdx.x * blockDim.x + threadIdx.x;
    const int nthreads = gridDim.x * blockDim.x;

    auto acc1 = [&](float p0, float p1, int lab, float u) {
        const int  pred     = (p1 > p0) ? 1 : 0;   // argmax, first index wins ties
        const bool accurate = (lab == pred);
        const bool certain  = (u <= th);
        const float t  = FAST_TANH(u);
        const float c  = accurate ? p1 : (1.0f - p1);
        const float w  = certain ? (1.0f - t) : t;
        const float cw = c * w;
        den += cw;
        num += (accurate == certain) ? cw : 0.0f;
    };

    // Main vectorized loop: 4 elements / thread / iteration.
    // Streamed-once data (268 MB > 192 MB L2) -> non-temporal loads.
    const int n4 = n >> 2;
    const f32x4* __restrict__ probs4 = (const f32x4*)probs;  // 2 x f32x4 per group of 4 elems
    const i32x4* __restrict__ lab4   = (const i32x4*)labels;
    const f32x4* __restrict__ unc4   = (const f32x4*)unc;

    for (int q = tid; q < n4; q += nthreads) {
        const f32x4 pA = __builtin_nontemporal_load(&probs4[2 * q]);
        const f32x4 pB = __builtin_nontemporal_load(&probs4[2 * q + 1]);
        const i32x4 lb = __builtin_nontemporal_load(&lab4[q]);
        const f32x4 uu = __builtin_nontemporal_load(&unc4[q]);
        acc1(pA.x, pA.y, lb.x, uu.x);
        acc1(pA.z, pA.w, lb.y, uu.y);
        acc1(pB.x, pB.y, lb.z, uu.z);
        acc1(pB.z, pB.w, lb.w, uu.w);
    }

    // Scalar tail (n % 4).
    for (int i = (n4 << 2) + tid; i < n; i += nthreads) {
        acc1(probs[2 * i], probs[2 * i + 1], labels[i], unc[i]);
    }

    // Wave32 reduction via WMMA (all 256 lanes reconverged here: EXEC all-1s).
    const float wnum = wave_reduce_sum_wmma(num);
    const float wden = wave_reduce_sum_wmma(den);

    // Block combine in LDS, then one global f32 atomic pair per block.
    __shared__ float s_num, s_den;
    if (threadIdx.x == 0) { s_num = 0.0f; s_den = 0.0f; }
    __syncthreads();
    if ((threadIdx.x & 31) == 0) {
        atomicAdd(&s_num, wnum);
        atomicAdd(&s_den, wden);
    }
    __syncthreads();
    if (threadIdx.x == 0) {
        atomicAdd(&ws[0], s_num);
        atomicAdd(&ws[1], s_den);
    }
}

__global__ void avu_final_kernel(const float* __restrict__ ws,
                                 float* __restrict__ out) {
    const float num = ws[0];
    const float den = ws[1];
    const float avu = num / (den + AVU_EPS);
    out[0] = -logf(avu + AVU_EPS);   // BETA = 1.0
}

extern "C" void kernel_launch(void* const* d_in, const int* in_sizes, int n_in,
                              void* d_out, int out_size, void* d_ws, size_t ws_size,
                              hipStream_t stream) {
    const float* probs  = (const float*)d_in[0];  // (N,2) f32
    const int*   labels = (const int*)d_in[1];    // (N,)  i32
    const float* unc    = (const float*)d_in[2];  // (N,)  f32
    const float* unc_th = (const float*)d_in[3];  // scalar f32
    float* out = (float*)d_out;
    float* ws  = (float*)d_ws;

    const int n = in_sizes[2];  // N (unc length)

    const int threads = 256;
    int blocks = (n + threads * 4 - 1) / (threads * 4);
    if (blocks > 2048) blocks = 2048;
    if (blocks < 1) blocks = 1;

    avu_ws_init_kernel<<<1, 1, 0, stream>>>(ws);
    avu_partial_kernel<<<blocks, threads, 0, stream>>>(probs, labels, unc, unc_th, ws, n);
    avu_final_kernel<<<1, 1, 0, stream>>>(ws, out);
}